// GATModel_18554258719429
// MI455X (gfx1250) — compile-verified
//
#include <hip/hip_runtime.h>

typedef float v2f __attribute__((ext_vector_type(2)));
typedef float v8f __attribute__((ext_vector_type(8)));

#define N_NODES 50000
#define N_EDGES 800000
#define N_GRAPH 100
#define HH 4
#define FF 32
#define HF 128          // H*F == DIN == 128
#define SLOPE 0.2f

static inline int cdiv(int a, int b) { return (a + b - 1) / b; }

// ---------------------------------------------------------------------------
// GEMM: Z[nrows,128] = A[nrows,128] @ W[128,128], fp32 WMMA 16x16x4.
// Block = 256 threads = 8 waves; each wave owns 16 rows x 128 cols (8 tiles).
// W is staged into LDS once per block (64 KB of the WGP's 320 KB).
// ---------------------------------------------------------------------------
__global__ __launch_bounds__(256)
void gemm128_wmma(const float* __restrict__ A, const float* __restrict__ W,
                  float* __restrict__ Z, int nrows) {
    __shared__ float Wl[HF * HF];
    for (int i = threadIdx.x; i < HF * HF; i += 256) Wl[i] = W[i];
    __syncthreads();

    const int wave = threadIdx.x >> 5;
    const int lane = threadIdx.x & 31;
    const int m    = lane & 15;            // row within tile / col within tile
    const int ksub = (lane >> 4) << 1;     // 0 or 2 : which K-pair this half-wave holds

    int row  = blockIdx.x * 128 + wave * 16 + m;
    int rowc = row < nrows ? row : (nrows - 1);      // clamp loads, keep EXEC full
    const float* arow = A + (size_t)rowc * HF;

    v8f c[8];
#pragma unroll
    for (int t = 0; t < 8; ++t) c[t] = (v8f){0.f,0.f,0.f,0.f,0.f,0.f,0.f,0.f};

    for (int k = 0; k < HF; k += 4) {
        // A 16x4 fragment: lane holds A[m, k+ksub] and A[m, k+ksub+1]
        v2f a = *(const v2f*)(arow + k + ksub);
#pragma unroll
        for (int t = 0; t < 8; ++t) {
            v2f b;
            b.x = Wl[(k + ksub)     * HF + t * 16 + m];
            b.y = Wl[(k + ksub + 1) * HF + t * 16 + m];
            c[t] = __builtin_amdgcn_wmma_f32_16x16x4_f32(
                false, a, false, b, (short)0, c[t], false, false);
        }
    }

    // C/D layout: VGPR r -> (M = r + (lane>>4)*8, N = lane&15)
    int rbase = blockIdx.x * 128 + wave * 16 + (lane >> 4) * 8;
    int col   = lane & 15;
#pragma unroll
    for (int r = 0; r < 8; ++r) {
        int orow = rbase + r;
        if (orow < nrows) {
#pragma unroll
            for (int t = 0; t < 8; ++t)
                Z[(size_t)orow * HF + t * 16 + col] = c[t][r];
        }
    }
}

// ---------------------------------------------------------------------------
// Attention coefficients: el[n,h] = sum_f z[n,h,f]*al[h,f]; er likewise.
// ---------------------------------------------------------------------------
__global__ __launch_bounds__(256)
void elr_kernel(const float* __restrict__ z, const float* __restrict__ al,
                const float* __restrict__ ar, float* __restrict__ el,
                float* __restrict__ er) {
    int idx = blockIdx.x * blockDim.x + threadIdx.x;   // n*H + h
    if (idx >= N_NODES * HH) return;
    int h = idx & 3;
    const float* zr = z + (size_t)(idx >> 2) * HF + h * FF;
    float sl = 0.f, sr = 0.f;
#pragma unroll
    for (int f = 0; f < FF; ++f) {
        float zv = zr[f];
        sl += zv * al[h * FF + f];
        sr += zv * ar[h * FF + f];
    }
    el[idx] = sl;
    er[idx] = sr;
}

// Monotone float<->uint encoding so unsigned atomicMax == float max.
__device__ __forceinline__ unsigned enc_f32(float f) {
    unsigned u = __float_as_uint(f);
    return (u & 0x80000000u) ? ~u : (u | 0x80000000u);
}
__device__ __forceinline__ float dec_f32(unsigned k) {
    return __uint_as_float((k & 0x80000000u) ? (k ^ 0x80000000u) : ~k);
}

__global__ __launch_bounds__(256)
void zero_f32(float* __restrict__ p, int n) {
    int i = blockIdx.x * blockDim.x + threadIdx.x;
    if (i < n) p[i] = 0.f;
}

// Pass A: e = leakyrelu(el[src]+er[dst]); segment max over dst (encoded atomicMax)
__global__ __launch_bounds__(256)
void edge_max(const int* __restrict__ src, const int* __restrict__ dst,
              const float* __restrict__ el, const float* __restrict__ er,
              float* __restrict__ ebuf, unsigned* __restrict__ menc) {
    int idx = blockIdx.x * blockDim.x + threadIdx.x;   // e*H + h
    if (idx >= N_EDGES * HH) return;
    int e = idx >> 2, h = idx & 3;
    int sd = src[e], dd = dst[e];
    float t = el[sd * HH + h] + er[dd * HH + h];
    t = t > 0.f ? t : SLOPE * t;
    ebuf[idx] = t;
    atomicMax(&menc[dd * HH + h], enc_f32(t));
}

// Pass B: p = exp(e - m[dst]); segment sum over dst  (p overwrites ebuf in place)
__global__ __launch_bounds__(256)
void edge_exp(const int* __restrict__ dst, const unsigned* __restrict__ menc,
              float* __restrict__ ebuf, float* __restrict__ s) {
    int idx = blockIdx.x * blockDim.x + threadIdx.x;   // e*H + h
    if (idx >= N_EDGES * HH) return;
    int e = idx >> 2, h = idx & 3;
    int dd = dst[e];
    float mval = dec_f32(menc[dd * HH + h]);
    float p = expf(ebuf[idx] - mval);
    ebuf[idx] = p;
    atomicAdd(&s[dd * HH + h], p);
}

// Pass C: out[dst] += (p/s[dst]) * z[src].  One wave32 per edge, 4 floats/lane.
__global__ __launch_bounds__(256)
void edge_aggr(const int* __restrict__ src, const int* __restrict__ dst,
               const float* __restrict__ p, const float* __restrict__ s,
               const float* __restrict__ z, float* __restrict__ out) {
    int gid  = blockIdx.x * blockDim.x + threadIdx.x;
    int e    = gid >> 5;
    int lane = gid & 31;
    if (e >= N_EDGES) return;
    int sd = src[e], dd = dst[e];
    int head = lane >> 3;                                // (lane*4)/32
    float alpha = p[e * HH + head] / s[dd * HH + head];
    const float4 zv = *(const float4*)(z + (size_t)sd * HF + lane * 4);
    float* op = out + (size_t)dd * HF + lane * 4;
    atomicAdd(op + 0, alpha * zv.x);
    atomicAdd(op + 1, alpha * zv.y);
    atomicAdd(op + 2, alpha * zv.z);
    atomicAdd(op + 3, alpha * zv.w);
}

__global__ __launch_bounds__(256)
void add_bias(const float* __restrict__ acc, const float* __restrict__ b,
              float* __restrict__ hnext) {
    int idx = blockIdx.x * blockDim.x + threadIdx.x;   // n*128 + i
    if (idx >= N_NODES * HF) return;
    hnext[idx] = acc[idx] + b[idx & (HF - 1)];
}

// Layer 3 readout: mean over heads, then per-graph sum via atomics.
__global__ __launch_bounds__(256)
void head_mean_scatter(const float* __restrict__ acc, const float* __restrict__ b,
                       const int* __restrict__ gids, float* __restrict__ gsum) {
    int idx = blockIdx.x * blockDim.x + threadIdx.x;   // n*F + f
    if (idx >= N_NODES * FF) return;
    int n = idx >> 5, f = idx & 31;
    float v = 0.f;
#pragma unroll
    for (int h = 0; h < HH; ++h)
        v += acc[(size_t)n * HF + h * FF + f] + b[h * FF + f];
    v *= (1.0f / HH);
    atomicAdd(&gsum[gids[n] * FF + f], v);
}

__global__ __launch_bounds__(256)
void count_nodes(const int* __restrict__ gids, float* __restrict__ gcnt) {
    int n = blockIdx.x * blockDim.x + threadIdx.x;
    if (n < N_NODES) atomicAdd(&gcnt[gids[n]], 1.0f);
}

__global__ __launch_bounds__(256)
void graph_mean(const float* __restrict__ gsum, const float* __restrict__ gcnt,
                float* __restrict__ out) {
    int idx = blockIdx.x * blockDim.x + threadIdx.x;   // g*F + f
    if (idx >= N_GRAPH * FF) return;
    out[idx] = gsum[idx] / fmaxf(gcnt[idx >> 5], 1.0f);
}

// ---------------------------------------------------------------------------
extern "C" void kernel_launch(void* const* d_in, const int* in_sizes, int n_in,
                              void* d_out, int out_size, void* d_ws, size_t ws_size,
                              hipStream_t stream) {
    (void)in_sizes; (void)n_in; (void)out_size; (void)ws_size;

    const float* x    = (const float*)d_in[0];
    const int*   src  = (const int*)d_in[1];
    const int*   dst  = (const int*)d_in[2];
    const int*   gids = (const int*)d_in[3];

    const int NHF = N_NODES * HF;     // 6.4M
    const int NH  = N_NODES * HH;     // 200k
    const int EH  = N_EDGES * HH;     // 3.2M

    float* ws   = (float*)d_ws;
    float* z    = ws;                 // [N,128]
    float* hbuf = z    + NHF;         // [N,128]
    float* oacc = hbuf + NHF;         // [N,128]
    float* ebuf = oacc + NHF;         // [E,H]
    float* el   = ebuf + EH;          // [N,H]
    float* er   = el   + NH;          // [N,H]
    float* s    = er   + NH;          // [N,H]
    unsigned* menc = (unsigned*)(s + NH);            // [N,H]
    float* gsum = (float*)(menc + NH);               // [G,F]
    float* gcnt = gsum + N_GRAPH * FF;               // [G]

    const int T = 256;
    const float* hin = x;

    for (int layer = 0; layer < 3; ++layer) {
        const float* W  = (const float*)d_in[4 + 4 * layer];
        const float* al = (const float*)d_in[5 + 4 * layer];
        const float* ar = (const float*)d_in[6 + 4 * layer];
        const float* bb = (const float*)d_in[7 + 4 * layer];

        gemm128_wmma<<<cdiv(N_NODES, 128), T, 0, stream>>>(hin, W, z, N_NODES);
        elr_kernel<<<cdiv(NH, T), T, 0, stream>>>(z, al, ar, el, er);

        zero_f32<<<cdiv(NH, T), T, 0, stream>>>((float*)menc, NH);   // 0 == encoded -lowest
        zero_f32<<<cdiv(NH, T), T, 0, stream>>>(s, NH);
        zero_f32<<<cdiv(NHF, T), T, 0, stream>>>(oacc, NHF);

        edge_max<<<cdiv(EH, T), T, 0, stream>>>(src, dst, el, er, ebuf, menc);
        edge_exp<<<cdiv(EH, T), T, 0, stream>>>(dst, menc, ebuf, s);
        edge_aggr<<<cdiv(N_EDGES * 32, T), T, 0, stream>>>(src, dst, ebuf, s, z, oacc);

        if (layer < 2) {
            add_bias<<<cdiv(NHF, T), T, 0, stream>>>(oacc, bb, hbuf);
            hin = hbuf;
        } else {
            zero_f32<<<cdiv(N_GRAPH * FF + N_GRAPH, T), T, 0, stream>>>(gsum, N_GRAPH * FF + N_GRAPH);
            head_mean_scatter<<<cdiv(N_NODES * FF, T), T, 0, stream>>>(oacc, bb, gids, gsum);
            count_nodes<<<cdiv(N_NODES, T), T, 0, stream>>>(gids, gcnt);
            graph_mean<<<cdiv(N_GRAPH * FF, T), T, 0, stream>>>(gsum, gcnt, (float*)d_out);
        }
    }
}